// yolo_loss_20761871909528
// MI455X (gfx1250) — compile-verified
//
#include <hip/hip_runtime.h>
#include <hip/hip_bf16.h>

typedef float v2f __attribute__((ext_vector_type(2)));
typedef float v8f __attribute__((ext_vector_type(8)));

#define B_      32
#define N_      10647
#define D_      85
#define C_      80
#define M_      50
#define T_      1600
#define NBLK_B  7          // ceil(1600/256)
#define NBLK_C  256
#define NPART   (NBLK_B + NBLK_C)
#define FLAGS_N (B_ * N_)          // 340704, divisible by 4
#define FLAGS_BYTES_PAD 340736     // 128B-aligned region for flags

// ---------------------------------------------------------------------------
// Wave-level f32 sum via V_WMMA_F32_16X16X4_F32.
// A(m,k): lane m<16 supplies k=0 (a[0]) and k=1 (a[1]); lanes 16-31 supply
// k=2,3. With a = {v, 0} and B = all-ones, D(m,n) = v[m] + v[m+16] for all n.
// Lane l holds D rows {0..7} (l<16) or {8..15} (l>=16) of column l%16, so
// sum of its 8 accumulator regs + xor-16 shuffle = full 32-lane sum, exact f32.
// EXEC must be all ones at this point (callers guarantee reconvergence).
// ---------------------------------------------------------------------------
__device__ __forceinline__ float wave_sum_wmma(float v) {
    v2f a; a[0] = v;    a[1] = 0.0f;
    v2f b; b[0] = 1.0f; b[1] = 1.0f;
    v8f c = {};
    c = __builtin_amdgcn_wmma_f32_16x16x4_f32(
            /*neg_a=*/false, a, /*neg_b=*/false, b,
            /*c_mod=*/(short)0, c, /*reuse_a=*/false, /*reuse_b=*/false);
    float s = c[0] + c[1] + c[2] + c[3] + c[4] + c[5] + c[6] + c[7];
    s += __shfl_xor(s, 16, 32);   // combine row-halves; all lanes get wave sum
    return s;
}

// Deterministic block sum (result valid on thread 0).
__device__ __forceinline__ float block_sum(float v) {
    __shared__ float swv[8];
    float ws = wave_sum_wmma(v);
    int lane = threadIdx.x & 31;
    int wv   = threadIdx.x >> 5;
    if (lane == 0) swv[wv] = ws;
    __syncthreads();
    float tot = 0.0f;
    if (threadIdx.x == 0) {
        int nw = (blockDim.x + 31) >> 5;
        for (int w = 0; w < nw; ++w) tot += swv[w];
    }
    __syncthreads();
    return tot;
}

// ---------------------------------------------------------------------------
// Kernel A: zero the flag array (re-run every call; harness never re-poisons).
// ---------------------------------------------------------------------------
__global__ void k_init(int* __restrict__ flags_i, int n_ints) {
    int i = blockIdx.x * blockDim.x + threadIdx.x;
    if (i < n_ints) flags_i[i] = 0;
}

// ---------------------------------------------------------------------------
// Kernel B: one thread per target. IoU argmax over 9 candidates, coord MSE,
// 80-class BCE, and flag the selected (img, sel) position.
// ---------------------------------------------------------------------------
__global__ void k_targets(const float* __restrict__ pred,
                          const float* __restrict__ coord_x,
                          const float* __restrict__ y_coord,
                          const int*   __restrict__ y_cls,
                          const float* __restrict__ anchors,
                          unsigned char* __restrict__ flags,
                          float* __restrict__ partials) {
    int t = blockIdx.x * blockDim.x + threadIdx.x;
    float partial = 0.0f;
    if (t < T_) {
        int b = t / M_;
        float gx0 = y_coord[t * 4 + 0];
        float gy0 = y_coord[t * 4 + 1];
        float gw  = y_coord[t * 4 + 2];
        float gh  = y_coord[t * 4 + 3];

        // GT corner box in input-pixel units
        float g1x = 416.0f * (gx0 - gw * 0.5f);
        float g1y = 416.0f * (gy0 - gh * 0.5f);
        float g2x = 416.0f * (gx0 + gw * 0.5f);
        float g2y = 416.0f * (gy0 + gh * 0.5f);
        float garea = (g2x - g1x) * (g2y - g1y);

        const float gridsf[3] = {13.0f, 26.0f, 52.0f};
        const int   basei[3]  = {0, 507, 2535};

        int cand[9];
        for (int gi = 0; gi < 3; ++gi) {
            float g  = gridsf[gi];
            float fx = floorf(gx0 * g);
            float fy = floorf(gy0 * g);
            int idx0 = basei[gi] + (int)((fy + fx * g) * 3.0f);  // cell = gy + gx*g
            cand[gi * 3 + 0] = idx0;
            cand[gi * 3 + 1] = idx0 + 1;
            cand[gi * 3 + 2] = idx0 + 2;
        }

        const float* prow_base = pred + (size_t)b * N_ * D_;
        int best_j = 0;
        float best_iou = -1.0f;
        for (int j = 0; j < 9; ++j) {
            const float* pr = prow_base + (size_t)cand[j] * D_;
            float cx = pr[0], cy = pr[1], w = pr[2], h = pr[3];
            float p1x = cx - w * 0.5f, p1y = cy - h * 0.5f;
            float p2x = cx + w * 0.5f, p2y = cy + h * 0.5f;
            float ix1 = fmaxf(g1x, p1x), iy1 = fmaxf(g1y, p1y);
            float ix2 = fminf(g2x, p2x), iy2 = fminf(g2y, p2y);
            float inter = fmaxf(ix2 - ix1, 0.0f) * fmaxf(iy2 - iy1, 0.0f);
            float pa = (p2x - p1x) * (p2y - p1y);
            float iou = inter / (garea + pa - inter + 1e-16f);
            if (iou > best_iou) { best_iou = iou; best_j = j; }   // first-max tie
        }

        int sel = cand[best_j];
        int gi = best_j / 3, ai = best_j % 3;
        float aw = anchors[(gi * 3 + ai) * 2 + 0];
        float ah = anchors[(gi * 3 + ai) * 2 + 1];
        float gs = gridsf[gi];

        float fx = gx0 * gs, fy = gy0 * gs;
        float mx = fx - floorf(fx) + 1e-5f;
        float my = fy - floorf(fy) + 1e-5f;
        float tx = logf(mx / (1.0f - mx));
        float ty = logf(my / (1.0f - my));
        float tw = logf(gw * 416.0f / aw);
        float th = logf(gh * 416.0f / ah);

        const float* cp = coord_x + ((size_t)b * N_ + sel) * 4;
        float d0 = cp[0] - tx, d1 = cp[1] - ty, d2 = cp[2] - tw, d3 = cp[3] - th;
        partial += d0 * d0 + d1 * d1 + d2 * d2 + d3 * d3;   // LAM_COORD = 1

        const float* pc = prow_base + (size_t)sel * D_ + 5;
        int k = y_cls[t];
        float cls = 0.0f;
        for (int c = 0; c < C_; ++c) {
            float p = pc[c];
            cls += (c == k) ? -logf(p) : -logf(1.0f - p);
        }
        partial += cls;

        flags[(size_t)b * N_ + sel] = 1;   // races benign: all write 1
    }
    float tot = block_sum(partial);
    if (threadIdx.x == 0) partials[blockIdx.x] = tot;
}

// ---------------------------------------------------------------------------
// Kernel C: dense obj/noobj BCE over B*N confidence values (340-byte stride).
// Memory-bound: ~44 MB effective traffic -> ~2us at 23.3 TB/s.
// ---------------------------------------------------------------------------
__global__ void k_conf(const float* __restrict__ pred,
                       const unsigned char* __restrict__ flags,
                       float* __restrict__ partials) {
    const int total  = B_ * N_;
    const int stride = gridDim.x * blockDim.x;
    float acc = 0.0f;
    for (int i = blockIdx.x * blockDim.x + threadIdx.x; i < total; i += stride) {
        // Hint the next grid-stride line into cache (global_prefetch_b8).
        __builtin_prefetch(&pred[(size_t)(i + stride) * D_ + 4], 0, 1);
        float p = pred[(size_t)i * D_ + 4];
        float v = flags[i] ? 5.0f * (-logf(p))          // LAM_OBJ   * obj_bce
                           : 0.5f * (-logf(1.0f - p));  // LAM_NOOBJ * noobj_bce
        acc += v;
    }
    float tot = block_sum(acc);
    if (threadIdx.x == 0) partials[blockIdx.x] = tot;
}

// ---------------------------------------------------------------------------
// Kernel D: deterministic final reduction of NPART partials -> scalar loss.
// ---------------------------------------------------------------------------
__global__ void k_final(const float* __restrict__ partials, float* __restrict__ out) {
    int tid = threadIdx.x;
    float v = 0.0f;
    if (tid < NPART)       v  = partials[tid];
    if (tid + 256 < NPART) v += partials[tid + 256];
    float tot = block_sum(v);
    if (tid == 0) out[0] = tot;
}

extern "C" void kernel_launch(void* const* d_in, const int* in_sizes, int n_in,
                              void* d_out, int out_size, void* d_ws, size_t ws_size,
                              hipStream_t stream) {
    const float* pred    = (const float*)d_in[0];   // (32, 10647, 85)
    const float* coord_x = (const float*)d_in[1];   // (32, 10647, 4)
    const float* y_coord = (const float*)d_in[2];   // (32, 50, 4)
    const int*   y_cls   = (const int*)d_in[3];     // (32, 50)
    const float* anchors = (const float*)d_in[4];   // (3, 3, 2)

    unsigned char* flags    = (unsigned char*)d_ws;
    float*         partials = (float*)((char*)d_ws + FLAGS_BYTES_PAD);
    float*         out      = (float*)d_out;

    const int n_ints = FLAGS_N / 4;   // 85176
    k_init   <<<(n_ints + 255) / 256, 256, 0, stream>>>((int*)d_ws, n_ints);
    k_targets<<<NBLK_B, 256, 0, stream>>>(pred, coord_x, y_coord, y_cls, anchors,
                                          flags, partials);
    k_conf   <<<NBLK_C, 256, 0, stream>>>(pred, flags, partials + NBLK_B);
    k_final  <<<1, 256, 0, stream>>>(partials, out);
}